// ChunkedSurpriseGatedSSD_38087769981207
// MI455X (gfx1250) — compile-verified
//
#include <hip/hip_runtime.h>
#include <hip/hip_bf16.h>

typedef __attribute__((ext_vector_type(2))) float v2f;
typedef __attribute__((ext_vector_type(8))) float v8f;
typedef __attribute__((ext_vector_type(4))) unsigned int v4u;
typedef __attribute__((ext_vector_type(8))) int v8i;
typedef __attribute__((ext_vector_type(4))) int v4i;

#define CHUNKSZ 64
#define NSTATE  128
#define PDIM    64
#define HHEADS  16
#define BSZ     2
#define SEQ     4096
#define NCHUNK  (SEQ / CHUNKSZ)

// LDS strides (padded: 64-dword rows + 2, 128-dword rows + 2 -> no bank conflicts,
// and exactly reproducible by the TDM pad_interval/pad_amount hardware padding)
#define XS_S 66
#define BC_S 130
#define MS_S 66
#define SMEM_FLOATS (64*XS_S + 64*BC_S + 64*BC_S + 64*MS_S + 3*64 + 256)

#define WMMA_F32(a, b, c) \
    __builtin_amdgcn_wmma_f32_16x16x4_f32(false, (a), false, (b), (short)0, (c), false, false)

#if defined(__HIP_DEVICE_COMPILE__) && __has_builtin(__builtin_amdgcn_tensor_load_to_lds)
#define USE_TDM 1
#else
#define USE_TDM 0
#endif

#if USE_TDM
// ---------------------------------------------------------------------------
// Tensor Data Mover: DMA a 2D tile (tile_w x tile_h elems, f32) from global to
// LDS, inserting pad dwords per row so the LDS image lands with a padded stride.
// pad codes: interval 5 = every 64 DW, 6 = every 128 DW; amount 1 = 2 DW pad.
// ---------------------------------------------------------------------------
__device__ __forceinline__ void tdm_load_2d_f32(unsigned int lds_addr, const void* gaddr,
                                                unsigned int width, unsigned int height,
                                                unsigned long long row_stride_elems,
                                                unsigned int pad_interval_code,
                                                unsigned int pad_amount_code)
{
    unsigned long long ga = (unsigned long long)(uintptr_t)gaddr;
    v4u g0;
    g0.x = 0x1u;                                            // count=1 (valid user D#)
    g0.y = lds_addr;                                        // LDS byte address
    g0.z = (unsigned int)ga;                                // global_addr[31:0]
    g0.w = (unsigned int)((ga >> 32) & 0x01FFFFFFu) | (2u << 30);  // addr[56:32], type=2
    v8i g1;
    g1[0] = (int)((2u << 16) | (1u << 20) |                 // data_size=4B, pad_enable
                  (pad_interval_code << 22) | (pad_amount_code << 25));
    g1[1] = (int)((width & 0xFFFFu) << 16);                 // tensor_dim0[15:0] (abar=0)
    g1[2] = (int)(((width >> 16) & 0xFFFFu) | ((height & 0xFFFFu) << 16)); // td0 hi | td1 lo
    g1[3] = (int)(((height >> 16) & 0xFFFFu) | ((width & 0xFFFFu) << 16)); // td1 hi | tile_dim0
    g1[4] = (int)(height & 0xFFFFu);                        // tile_dim1 (tile_dim2 = 0)
    g1[5] = (int)(unsigned int)row_stride_elems;            // tensor_dim0_stride[31:0]
    g1[6] = (int)(unsigned int)((row_stride_elems >> 32) & 0xFFFFu); // stride hi (dim1_stride=0)
    g1[7] = 0;
    v4i z4 = {0, 0, 0, 0};
#if __has_include(<hip/amd_detail/amd_gfx1250_TDM.h>)
    v8i z8 = {0, 0, 0, 0, 0, 0, 0, 0};
    __builtin_amdgcn_tensor_load_to_lds(g0, g1, z4, z4, z8, 0);
#else
    __builtin_amdgcn_tensor_load_to_lds(g0, g1, z4, z4, 0);
#endif
}
#endif

// ---------------------------------------------------------------------------
// Gate kernel: one wave32. EMA / alpha / scalar decay_scale for chunk c.
// ---------------------------------------------------------------------------
__global__ __launch_bounds__(32)
void surprise_gate_kernel(const float* __restrict__ l2ab,
                          const float* __restrict__ l2b,
                          const float* __restrict__ ema_in,
                          const float* __restrict__ wErr,
                          float* __restrict__ wEma,
                          float* __restrict__ wDs,
                          int c)
{
    int t = threadIdx.x;             // (b = t>>4, h = t&15)
    if (c == 0) {
        if (t == 0) wDs[0] = 1.0f;
        if (t < HHEADS) wEma[t] = ema_in[t];
        return;
    }
    int h = t & 15;
    float err       = wErr[t];
    float err_other = __shfl_xor(err, 16);
    float err_mean  = 0.5f * (err + err_other);
    float ema_new   = 0.99f * wEma[h] + 0.01f * err_mean;

    float beta = exp2f(fminf(fmaxf(l2b[h], -2.0f), 2.0f));
    float ab   = 1.0f - exp2f(fminf(fmaxf(l2ab[h], -3.32f), -0.015f));
    float norm = err / (ema_new + 1e-6f);
    float boost = fmaxf(0.0f, tanhf(beta * norm));
    float alpha = fminf(fmaxf(ab + (1.0f - ab) * boost, 0.01f), 0.999f);
    float om = 1.0f - alpha;
    #pragma unroll
    for (int off = 16; off > 0; off >>= 1) om += __shfl_xor(om, off);
    if (t == 0) wDs[0] = om * (1.0f / 32.0f);
    if (t < HHEADS) wEma[t] = ema_new;
}

// ---------------------------------------------------------------------------
// Chunk kernel: one block per (b,h), 256 threads = 8 waves.
// TDM stages X/B/C tiles into LDS (overlapped with the A cumsum), then all
// GEMMs run on V_WMMA_F32_16X16X4_F32 with shared A-fragments feeding 2-4
// independent accumulator chains per wave.
// ---------------------------------------------------------------------------
__global__ __launch_bounds__(256, 1)
void ssd_chunk_kernel(const float* __restrict__ X,
                      const float* __restrict__ A,
                      const float* __restrict__ Bm,
                      const float* __restrict__ Cm,
                      float* __restrict__ Y,
                      const float* __restrict__ hIn,
                      float* __restrict__ hOut,
                      float* __restrict__ wErr,
                      const float* __restrict__ wDs,
                      int c)
{
    extern __shared__ float smem[];
    float* Xs   = smem;                 // [64][66]
    float* Bs   = Xs + 64 * XS_S;       // [64][130]
    float* Cs   = Bs + 64 * BC_S;       // [64][130]
    float* Ms   = Cs + 64 * BC_S;       // [64][66]  (L .* (C B^T))
    float* sAcs = Ms + 64 * MS_S;       // [64]
    float* sDfs = sAcs + 64;            // [64] exp(Acs)
    float* sDte = sDfs + 64;            // [64] exp(Acs[63]-Acs)
    float* sRed = sDte + 64;            // [256]

    const int tid  = threadIdx.x;
    const int blk  = blockIdx.x;        // b*H + h
    const int b    = blk >> 4;
    const int h    = blk & 15;
    const int lane = tid & 31;
    const int wave = tid >> 5;
    const int l15  = lane & 15;
    const int half = lane >> 4;         // K-half selector for wmma fragments

    const float dscale = wDs[0];

    const size_t baseX = ((size_t)(b * SEQ + c * CHUNKSZ) * HHEADS + h) * PDIM;   // row stride 1024
    const size_t baseB = ((size_t)(b * SEQ + c * CHUNKSZ) * HHEADS + h) * NSTATE; // row stride 2048
    const size_t baseA = (size_t)(b * SEQ + c * CHUNKSZ) * HHEADS + h;            // stride 16
    const size_t baseH = (size_t)blk * (NSTATE * PDIM);

#if USE_TDM
    // ---- TDM staging: issue 3 tile DMAs from wave 0, overlap A cumsum ----
    if (wave == 0) {
        tdm_load_2d_f32((unsigned int)(uintptr_t)(void*)Xs, X + baseX,
                        PDIM, CHUNKSZ, HHEADS * PDIM, 5u /*64DW*/, 1u /*+2DW*/);
        tdm_load_2d_f32((unsigned int)(uintptr_t)(void*)Bs, Bm + baseB,
                        NSTATE, CHUNKSZ, HHEADS * NSTATE, 6u /*128DW*/, 1u);
        tdm_load_2d_f32((unsigned int)(uintptr_t)(void*)Cs, Cm + baseB,
                        NSTATE, CHUNKSZ, HHEADS * NSTATE, 6u, 1u);
    }
    if (tid < 64) sAcs[tid] = dscale * A[baseA + (size_t)tid * HHEADS];
    __syncthreads();
    if (tid == 0) {
        float run = 0.0f;
        for (int t = 0; t < 64; ++t) { run += sAcs[t]; sAcs[t] = run; }
    }
    __syncthreads();
    if (tid < 64) {
        sDfs[tid] = expf(sAcs[tid]);
        sDte[tid] = expf(sAcs[63] - sAcs[tid]);
    }
    if (wave == 0) __builtin_amdgcn_s_wait_tensorcnt(0);
    __syncthreads();   // X/B/C tiles + decay vectors now valid for everyone
#else
    // ---- fallback: manual float4 staging ----
    #pragma unroll
    for (int i = 0; i < 4; ++i) {
        int e = tid + i * 256;
        int t = e >> 4, c4 = e & 15;
        float4 v = ((const float4*)(X + baseX + (size_t)t * (HHEADS * PDIM)))[c4];
        float* d = &Xs[t * XS_S + c4 * 4];
        d[0] = v.x; d[1] = v.y; d[2] = v.z; d[3] = v.w;
    }
    #pragma unroll
    for (int i = 0; i < 8; ++i) {
        int e = tid + i * 256;
        int t = e >> 5, c4 = e & 31;
        float4 vb = ((const float4*)(Bm + baseB + (size_t)t * (HHEADS * NSTATE)))[c4];
        float* db = &Bs[t * BC_S + c4 * 4];
        db[0] = vb.x; db[1] = vb.y; db[2] = vb.z; db[3] = vb.w;
        float4 vc = ((const float4*)(Cm + baseB + (size_t)t * (HHEADS * NSTATE)))[c4];
        float* dc = &Cs[t * BC_S + c4 * 4];
        dc[0] = vc.x; dc[1] = vc.y; dc[2] = vc.z; dc[3] = vc.w;
    }
    if (tid < 64) sAcs[tid] = dscale * A[baseA + (size_t)tid * HHEADS];
    __syncthreads();
    if (tid == 0) {
        float run = 0.0f;
        for (int t = 0; t < 64; ++t) { run += sAcs[t]; sAcs[t] = run; }
    }
    __syncthreads();
    if (tid < 64) {
        sDfs[tid] = expf(sAcs[tid]);
        sDte[tid] = expf(sAcs[63] - sAcs[tid]);
    }
    __syncthreads();
#endif

    const v8f Z8 = {0.f, 0.f, 0.f, 0.f, 0.f, 0.f, 0.f, 0.f};

    // ========= phase 1: CB = C * B^T, masked/scaled into Ms ================
    // wave -> row-tile ti = wave>>1, col-tiles {tj0, tj0+1}; shared A-fragment.
    {
        const int ti   = wave >> 1;
        const int tj0  = (wave & 1) * 2;
        const int rowC = ti * 16 + l15;
        const int colB = tj0 * 16 + l15;
        v8f acc0 = Z8, acc1 = Z8;
        #pragma unroll 8
        for (int k0 = 0; k0 < NSTATE; k0 += 4) {
            int ka = k0 + 2 * half;
            v2f a, b0, b1;
            a.x  = Cs[rowC * BC_S + ka];
            a.y  = Cs[rowC * BC_S + ka + 1];
            b0.x = Bs[colB * BC_S + ka];            // B^T: [k][j] = Bs[j][k]
            b0.y = Bs[colB * BC_S + ka + 1];
            b1.x = Bs[(colB + 16) * BC_S + ka];
            b1.y = Bs[(colB + 16) * BC_S + ka + 1];
            acc0 = WMMA_F32(a, b0, acc0);
            acc1 = WMMA_F32(a, b1, acc1);
        }
        float acsj0 = sAcs[tj0 * 16 + l15];
        float acsj1 = sAcs[tj0 * 16 + 16 + l15];
        #pragma unroll
        for (int r = 0; r < 8; ++r) {
            int i  = ti * 16 + r + 8 * half;
            int j0 = tj0 * 16 + l15;
            Ms[i * MS_S + j0]      = (i >= j0)      ? expf(sAcs[i] - acsj0) * acc0[r] : 0.0f;
            Ms[i * MS_S + j0 + 16] = (i >= j0 + 16) ? expf(sAcs[i] - acsj1) * acc1[r] : 0.0f;
        }
    }
    __syncthreads();

    // ===== phase 2: Y = Ms*X + diag(dfs) * (C * h_prev)  (64x64 out) =======
    {
        const int ti   = wave >> 1;
        const int pj0  = (wave & 1) * 2;
        const int rowI = ti * 16 + l15;
        const int colP = pj0 * 16 + l15;
        v8f acc0 = Z8, acc1 = Z8;
        #pragma unroll 4
        for (int k0 = 0; k0 < CHUNKSZ; k0 += 4) {       // intra-chunk
            int ka = k0 + 2 * half;
            v2f a, b0, b1;
            a.x  = Ms[rowI * MS_S + ka];
            a.y  = Ms[rowI * MS_S + ka + 1];
            b0.x = Xs[ka * XS_S + colP];
            b0.y = Xs[(ka + 1) * XS_S + colP];
            b1.x = Xs[ka * XS_S + colP + 16];
            b1.y = Xs[(ka + 1) * XS_S + colP + 16];
            acc0 = WMMA_F32(a, b0, acc0);
            acc1 = WMMA_F32(a, b1, acc1);
        }
        if (c != 0) {                                   // inter-chunk (h_prev)
            float dfsr = sDfs[rowI];                    // row scale folded into A-frag
            #pragma unroll 4
            for (int k0 = 0; k0 < NSTATE; k0 += 4) {
                int ka = k0 + 2 * half;
                v2f a, b0, b1;
                a.x  = dfsr * Cs[rowI * BC_S + ka];
                a.y  = dfsr * Cs[rowI * BC_S + ka + 1];
                b0.x = hIn[baseH + (size_t)ka * PDIM + colP];
                b0.y = hIn[baseH + (size_t)(ka + 1) * PDIM + colP];
                b1.x = hIn[baseH + (size_t)ka * PDIM + colP + 16];
                b1.y = hIn[baseH + (size_t)(ka + 1) * PDIM + colP + 16];
                acc0 = WMMA_F32(a, b0, acc0);
                acc1 = WMMA_F32(a, b1, acc1);
            }
        }
        #pragma unroll
        for (int r = 0; r < 8; ++r) {
            int i = ti * 16 + r + 8 * half;
            Y[baseX + (size_t)i * (HHEADS * PDIM) + colP]      = acc0[r];
            Y[baseX + (size_t)i * (HHEADS * PDIM) + colP + 16] = acc1[r];
        }
    }

    // === phase 3: h_contrib = (B .* dte)^T * X ; h_end ; surprise err ======
    // wave -> N-tile ni = wave; all four P-tiles share the A-fragment.
    const float dtot = expf(sAcs[63]);
    float sq = 0.0f;
    {
        const int rowN = wave * 16 + l15;
        v8f acc[4] = {Z8, Z8, Z8, Z8};
        #pragma unroll 4
        for (int k0 = 0; k0 < CHUNKSZ; k0 += 4) {
            int ka = k0 + 2 * half;
            v2f a;
            a.x = sDte[ka]     * Bs[ka * BC_S + rowN];        // (B^T scaled by dte[k])
            a.y = sDte[ka + 1] * Bs[(ka + 1) * BC_S + rowN];
            #pragma unroll
            for (int j = 0; j < 4; ++j) {
                v2f bb;
                bb.x = Xs[ka * XS_S + j * 16 + l15];
                bb.y = Xs[(ka + 1) * XS_S + j * 16 + l15];
                acc[j] = WMMA_F32(a, bb, acc[j]);
            }
        }
        #pragma unroll
        for (int j = 0; j < 4; ++j) {
            #pragma unroll
            for (int r = 0; r < 8; ++r) {
                int n = wave * 16 + r + 8 * half;
                size_t g = baseH + (size_t)n * PDIM + j * 16 + l15;
                float hc = acc[j][r];
                float hp = (c != 0) ? hIn[g] : 0.0f;
                hOut[g] = dtot * hp + hc;
                sq += hc * hc;   // h_end - decay_total*h_prev == h_contrib exactly
            }
        }
    }
    sRed[tid] = sq;
    __syncthreads();
    if (tid == 0) {
        float s = 0.0f;
        for (int i = 0; i < 256; ++i) s += sRed[i];
        wErr[blk] = s * (1.0f / (NSTATE * PDIM));   // surprise err for chunk c+1
    }
}

// ---------------------------------------------------------------------------
extern "C" void kernel_launch(void* const* d_in, const int* in_sizes, int n_in,
                              void* d_out, int out_size, void* d_ws, size_t ws_size,
                              hipStream_t stream) {
    (void)in_sizes; (void)n_in; (void)out_size; (void)ws_size;
    const float* X    = (const float*)d_in[0];
    const float* A    = (const float*)d_in[1];
    const float* Bm   = (const float*)d_in[2];
    const float* Cm   = (const float*)d_in[3];
    const float* l2ab = (const float*)d_in[4];
    const float* l2b  = (const float*)d_in[5];
    const float* ema  = (const float*)d_in[6];
    float* Yout = (float*)d_out;

    // workspace: two h ping-pong buffers + err[32] + ema[16] + decay_scale[1]
    float* ws   = (float*)d_ws;
    float* h0   = ws;
    float* h1   = h0 + (size_t)BSZ * HHEADS * NSTATE * PDIM;   // 262144 floats each
    float* werr = h1 + (size_t)BSZ * HHEADS * NSTATE * PDIM;
    float* wema = werr + 32;
    float* wds  = wema + 16;

    const size_t smem = SMEM_FLOATS * sizeof(float);   // ~102 KB (< 320 KB/WGP)
    (void)hipFuncSetAttribute((const void*)ssd_chunk_kernel,
                              hipFuncAttributeMaxDynamicSharedMemorySize, (int)smem);

    for (int c = 0; c < NCHUNK; ++c) {
        surprise_gate_kernel<<<1, 32, 0, stream>>>(l2ab, l2b, ema, werr, wema, wds, c);
        const float* hin  = (c & 1) ? h1 : h0;
        float*       hout = (c & 1) ? h0 : h1;
        ssd_chunk_kernel<<<BSZ * HHEADS, 256, smem, stream>>>(
            X, A, Bm, Cm, Yout, hin, hout, werr, wds, c);
    }
}